// DeBERTaPositionalEmbeddings_57552561766508
// MI455X (gfx1250) — compile-verified
//
#include <hip/hip_runtime.h>

typedef float v2f __attribute__((ext_vector_type(2)));
typedef float v4f __attribute__((ext_vector_type(4)));
typedef float v8f __attribute__((ext_vector_type(8)));

#define NB   2
#define NS   2048
#define NH   16
#define NDH  64
#define NDIM 1024
#define NWIN 512
#define LNEPS 1e-5f

static __device__ __forceinline__ v8f wmma_f32(v2f a, v2f b, v8f c) {
    // D = A(16x4 f32) * B(4x16 f32) + C(16x16 f32)
    return __builtin_amdgcn_wmma_f32_16x16x4_f32(
        /*neg_a=*/false, a, /*neg_b=*/false, b,
        /*c_mod=*/(short)0, c, /*reuse_a=*/false, /*reuse_b=*/false);
}

// ---------------- Kernel 1: LayerNorm of relpos_emb [512,1024] -> pe (ws) ----
__global__ void __launch_bounds__(256) k_layernorm(
        const float* __restrict__ x, const float* __restrict__ g,
        const float* __restrict__ bta, float* __restrict__ pe) {
    const int row = blockIdx.x;
    const int tid = threadIdx.x;
    const float* xr = x + (size_t)row * NDIM;
    float v[4], s = 0.f, s2 = 0.f;
#pragma unroll
    for (int p = 0; p < 4; ++p) {
        v[p] = xr[tid + 256 * p];
        s += v[p];
        s2 += v[p] * v[p];
    }
    __shared__ float sa[256], sb[256];
    sa[tid] = s; sb[tid] = s2;
    __syncthreads();
    for (int off = 128; off > 0; off >>= 1) {
        if (tid < off) { sa[tid] += sa[tid + off]; sb[tid] += sb[tid + off]; }
        __syncthreads();
    }
    const float mu  = sa[0] * (1.0f / NDIM);
    const float var = sb[0] * (1.0f / NDIM) - mu * mu;
    const float rs  = rsqrtf(var + LNEPS);
#pragma unroll
    for (int p = 0; p < 4; ++p) {
        const int c = tid + 256 * p;
        pe[(size_t)row * NDIM + c] = (v[p] - mu) * rs * g[c] + bta[c];
    }
}

// ------- Kernel 2: relpos_{q,k} = pe @ W{q,k}^T + b  (512x1024 x 1024x1024) --
// One wave per 16x16 tile; 8 waves/block. 4096 tiles total (2 matrices).
__global__ void __launch_bounds__(256) k_proj(
        const float* __restrict__ pe,
        const float* __restrict__ Wq, const float* __restrict__ bq,
        const float* __restrict__ Wk, const float* __restrict__ bk,
        float* __restrict__ relq, float* __restrict__ relk) {
    const int lane = threadIdx.x;
    const int r    = lane & 15;
    const int half = lane >> 4;
    const int tile = blockIdx.x * 8 + threadIdx.y;
    const int mat  = tile >> 11;           // 0 = Wq path, 1 = Wk path
    const int rem  = tile & 2047;
    const int mt   = rem >> 6;             // 0..31  (M = 512)
    const int nt   = rem & 63;             // 0..63  (N = 1024)
    const float* W   = mat ? Wk : Wq;
    const float* bb  = mat ? bk : bq;
    float*       dst = mat ? relk : relq;

    const float* arow = pe + (size_t)(mt * 16 + r) * NDIM;
    const float* brow = W  + (size_t)(nt * 16 + r) * NDIM;   // B[k,n] = W[n,k]

    v8f c = {};
#pragma unroll 4
    for (int kk = 0; kk < NDIM / 4; ++kk) {
        const int k = kk * 4 + 2 * half;
        v2f a  = *(const v2f*)(arow + k);
        v2f bf = *(const v2f*)(brow + k);
        c = wmma_f32(a, bf, c);
    }
    const float bias = bb[nt * 16 + r];
#pragma unroll
    for (int v = 0; v < 8; ++v)
        dst[(size_t)(mt * 16 + v + 8 * half) * NDIM + nt * 16 + r] = c[v] + bias;
}

// --- Kernel 3: fused q@relpos_k^T scores (16x512 in LDS) + relpos gather -----
// Block = (b, h, strip of 16 q-rows). 8 waves x 4 win-tiles each.
__global__ void __launch_bounds__(256) k_query_pos(
        const float* __restrict__ q,
        const float* __restrict__ rk,   // relpos_k [512,1024]
        float* __restrict__ out1) {     // logits_query_pos
    __shared__ float sc[16 * NWIN];                          // 32 KB
    const int lane = threadIdx.x;
    const int r    = lane & 15;
    const int half = lane >> 4;
    const int it   = blockIdx.x & 127;
    const int bh   = blockIdx.x >> 7;
    const int h    = bh & 15;
    const int b    = bh >> 4;
    const int i0   = it * 16;
    const int wt0  = threadIdx.y * 4;

    const float* qrow = q  + ((size_t)(b * NS + i0 + r) * NH + h) * NDH;
    const float* rkh  = rk + (size_t)h * NDH;

    v8f acc[4] = {};
#pragma unroll
    for (int kk = 0; kk < NDH / 4; ++kk) {
        const int k = kk * 4 + 2 * half;
        v2f a = __builtin_nontemporal_load((const v2f*)(qrow + k)); // read-once
#pragma unroll
        for (int t = 0; t < 4; ++t) {
            const int w = (wt0 + t) * 16 + r;
            v2f bf = *(const v2f*)(rkh + (size_t)w * NDIM + k);     // L2-resident
            acc[t] = wmma_f32(a, bf, acc[t]);
        }
    }
#pragma unroll
    for (int t = 0; t < 4; ++t)
#pragma unroll
        for (int v = 0; v < 8; ++v)
            sc[(v + 8 * half) * NWIN + (wt0 + t) * 16 + r] = acc[t][v];
    __syncthreads();

    // Expand: out[b,h,i,j] = sc[i-i0, clip(j-i,-511,0)+511]; streamed b128 NT.
    const int tid = threadIdx.y * 32 + threadIdx.x;
    float* obase = out1 + ((size_t)(b * NH + h) * NS + i0) * NS;
#pragma unroll
    for (int e = tid; e < (16 * NS) / 4; e += 256) {
        const int idx = e * 4;
        const int row = idx >> 11;
        const int jv  = idx & (NS - 1);
        const int i   = i0 + row;
        v4f o;
#pragma unroll
        for (int u = 0; u < 4; ++u) {
            int d = jv + u - i;
            d = d < -(NWIN - 1) ? -(NWIN - 1) : (d > 0 ? 0 : d);
            o[u] = sc[row * NWIN + d + (NWIN - 1)];
        }
        __builtin_nontemporal_store(o, (v4f*)(obase + (size_t)row * NS + jv));
    }
}

// --- Kernel 4: fused relpos_q@k^T scores (512x16 in LDS) + relpos gather -----
// Block = (b, h, strip of 16 k-columns). 8 waves x 4 w-tiles each.
__global__ void __launch_bounds__(256) k_pos_key(
        const float* __restrict__ kin,
        const float* __restrict__ rq,     // relpos_q [512,1024]
        float* __restrict__ out0) {       // logits_pos_key
    __shared__ float sc[NWIN * 16];                          // 32 KB
    const int lane = threadIdx.x;
    const int r    = lane & 15;
    const int half = lane >> 4;
    const int jt   = blockIdx.x & 127;
    const int bh   = blockIdx.x >> 7;
    const int h    = bh & 15;
    const int b    = bh >> 4;
    const int j0   = jt * 16;
    const int mt0  = threadIdx.y * 4;

    const float* krow = kin + ((size_t)(b * NS + j0 + r) * NH + h) * NDH;
    const float* rqh  = rq  + (size_t)h * NDH;

    v8f acc[4] = {};
#pragma unroll
    for (int kk = 0; kk < NDH / 4; ++kk) {
        const int k = kk * 4 + 2 * half;
        v2f bf = __builtin_nontemporal_load((const v2f*)(krow + k)); // read-once
#pragma unroll
        for (int t = 0; t < 4; ++t) {
            const int w = (mt0 + t) * 16 + r;
            v2f a = *(const v2f*)(rqh + (size_t)w * NDIM + k);       // L2-resident
            acc[t] = wmma_f32(a, bf, acc[t]);
        }
    }
#pragma unroll
    for (int t = 0; t < 4; ++t)
#pragma unroll
        for (int v = 0; v < 8; ++v)
            sc[((mt0 + t) * 16 + v + 8 * half) * 16 + r] = acc[t][v];
    __syncthreads();

    // Expand: out[b,h,i,j] = sc[clip(j-i,-511,0)+511, j-j0]; streamed b128 NT.
    const int tid = threadIdx.y * 32 + threadIdx.x;
    float* obase = out0 + (size_t)(b * NH + h) * NS * NS + j0;
#pragma unroll
    for (int e = tid; e < (NS * 16) / 4; e += 256) {
        const int idx = e * 4;
        const int i   = idx >> 4;
        const int jj  = idx & 15;
        v4f o;
#pragma unroll
        for (int u = 0; u < 4; ++u) {
            int d = j0 + jj + u - i;
            d = d < -(NWIN - 1) ? -(NWIN - 1) : (d > 0 ? 0 : d);
            o[u] = sc[(d + NWIN - 1) * 16 + jj + u];
        }
        __builtin_nontemporal_store(o, (v4f*)(obase + (size_t)i * NS + jj));
    }
}

extern "C" void kernel_launch(void* const* d_in, const int* in_sizes, int n_in,
                              void* d_out, int out_size, void* d_ws, size_t ws_size,
                              hipStream_t stream) {
    const float* q   = (const float*)d_in[0];
    const float* k   = (const float*)d_in[1];
    const float* rpe = (const float*)d_in[2];
    const float* lns = (const float*)d_in[3];
    const float* lnb = (const float*)d_in[4];
    const float* Wq  = (const float*)d_in[5];
    const float* bq  = (const float*)d_in[6];
    const float* Wk  = (const float*)d_in[7];
    const float* bk  = (const float*)d_in[8];

    float* out = (float*)d_out;
    const size_t L = (size_t)NB * NH * NS * NS;      // 134,217,728
    float* out0 = out;                               // logits_pos_key
    float* out1 = out + L;                           // logits_query_pos
    float* relq = out + 2 * L;                       // relpos_q [512,16,64]
    float* relk = out + 2 * L + (size_t)NWIN * NDIM; // relpos_k [512,16,64]

    float* pe = (float*)d_ws;                        // [512,1024] = 2 MB

    k_layernorm<<<NWIN, 256, 0, stream>>>(rpe, lns, lnb, pe);
    k_proj<<<512, dim3(32, 8), 0, stream>>>(pe, Wq, bq, Wk, bk, relq, relk);
    k_query_pos<<<NB * NH * (NS / 16), dim3(32, 8), 0, stream>>>(q, relk, out1);
    k_pos_key  <<<NB * NH * (NS / 16), dim3(32, 8), 0, stream>>>(k, relq, out0);
}